// HANLayer_62105227100408
// MI455X (gfx1250) — compile-verified
//
#include <hip/hip_runtime.h>
#include <math.h>

#define N_NODES 50000
#define E_EDGES 800000
#define IN_DIM  128
#define HEADS   4
#define DHEAD   32
#define HD      128          // HEADS*DHEAD
#define MPATHS  3
#define HID_DIM 128
#define SLOPE   0.2f

typedef __attribute__((ext_vector_type(2))) float v2f;
typedef __attribute__((ext_vector_type(8))) float v8f;

// ---- order-preserving float<->uint encoding for atomicMax on floats ----
__device__ __forceinline__ unsigned enc_f(float f) {
  unsigned u = __float_as_uint(f);
  return (u & 0x80000000u) ? ~u : (u | 0x80000000u);
}
__device__ __forceinline__ float dec_f(unsigned u) {
  return __uint_as_float((u & 0x80000000u) ? (u & 0x7FFFFFFFu) : ~u);
}

// ================= Kernel 1: feat = h @ W via V_WMMA_F32_16X16X4_F32 ======
// grid.x = N/16 row blocks; 8 waves per block, wave w owns column tile w.
__global__ void feat_gemm(const float* __restrict__ h,
                          const float* __restrict__ W,
                          float* __restrict__ feat) {
  const int wave = threadIdx.x >> 5;     // 0..7 -> 16-col tile
  const int lane = threadIdx.x & 31;
  const int r0 = blockIdx.x * 16;
  const int c0 = wave * 16;
  const int hf = lane >> 4;              // lane half: K offset selector
  const int l  = lane & 15;

  v8f acc = {0.f,0.f,0.f,0.f,0.f,0.f,0.f,0.f};
  const float* arow = h + (size_t)(r0 + l) * IN_DIM;
  for (int k0 = 0; k0 < IN_DIM; k0 += 4) {
    v2f a, b;
    // A 16x4 f32 frag: lanes0-15 -> K0,K1 ; lanes16-31 -> K2,K3
    a.x = arow[k0 + hf * 2 + 0];
    a.y = arow[k0 + hf * 2 + 1];
    // B 4x16 frag mirrored: per-VGPR one K-row striped over 16 lanes
    b.x = W[(size_t)(k0 + hf * 2 + 0) * HD + c0 + l];
    b.y = W[(size_t)(k0 + hf * 2 + 1) * HD + c0 + l];
    acc = __builtin_amdgcn_wmma_f32_16x16x4_f32(false, a, false, b,
                                                (short)0, acc, false, false);
  }
  // C/D layout: VGPR r, lanes0-15 -> row r0+r ; lanes16-31 -> row r0+8+r
  #pragma unroll
  for (int r = 0; r < 8; ++r)
    feat[(size_t)(r0 + hf * 8 + r) * HD + c0 + l] = acc[r];
}

// ================= Kernel 2: el/er = (feat * attn)[per head] ==============
__global__ void el_er_kernel(const float* __restrict__ feat,
                             const float* __restrict__ attn_l,
                             const float* __restrict__ attn_r,
                             float* __restrict__ el, float* __restrict__ er) {
  int idx = blockIdx.x * blockDim.x + threadIdx.x;   // n*H + h
  if (idx >= N_NODES * HEADS) return;
  int n = idx >> 2, hh = idx & 3;
  const float* f  = feat + (size_t)n * HD + hh * DHEAD;
  const float* al = attn_l + hh * DHEAD;
  const float* ar = attn_r + hh * DHEAD;
  float sl = 0.f, sr = 0.f;
  #pragma unroll
  for (int d = 0; d < DHEAD; ++d) { float v = f[d]; sl += v * al[d]; sr += v * ar[d]; }
  el[idx] = sl; er[idx] = sr;
}

// ================= per-metapath init ======================================
__global__ void init_mp(unsigned* __restrict__ m_enc, float* __restrict__ denom,
                        float* __restrict__ z, int mp) {
  int idx = blockIdx.x * blockDim.x + threadIdx.x;
  if (idx < N_NODES * HD) {
    int n = idx >> 7, c = idx & 127;
    z[(size_t)n * (MPATHS * HD) + mp * HD + c] = 0.f;
  }
  if (idx < N_NODES * HEADS) { m_enc[idx] = 0u; denom[idx] = 0.f; }
}

__global__ void zero_score(float* score) {
  if (threadIdx.x < MPATHS) score[threadIdx.x] = 0.f;
}

// ================= Kernel 3: edge logits + per-dst max ====================
__global__ void edge_logits_max(const int* __restrict__ src, const int* __restrict__ dst,
                                const float* __restrict__ el, const float* __restrict__ er,
                                float* __restrict__ e_buf, unsigned* __restrict__ m_enc) {
  int idx = blockIdx.x * blockDim.x + threadIdx.x;   // edge*H + h
  if (idx >= E_EDGES * HEADS) return;
  int eid = idx >> 2, hh = idx & 3;
  int s = src[eid], d = dst[eid];
  float e = el[s * 4 + hh] + er[d * 4 + hh];
  e = e > 0.f ? e : SLOPE * e;
  e_buf[idx] = e;
  atomicMax(m_enc + d * 4 + hh, enc_f(e));
}

// ================= Kernel 4: softmax denominators =========================
__global__ void edge_denom(const int* __restrict__ dst, const float* __restrict__ e_buf,
                           const unsigned* __restrict__ m_enc, float* __restrict__ denom) {
  int idx = blockIdx.x * blockDim.x + threadIdx.x;
  if (idx >= E_EDGES * HEADS) return;
  int eid = idx >> 2, hh = idx & 3;
  int d = dst[eid];
  float a = __expf(e_buf[idx] - dec_f(m_enc[d * 4 + hh]));
  atomicAdd(denom + d * 4 + hh, a);
}

// ================= Kernel 5: aggregate (1 wave per edge) ==================
__global__ void edge_aggregate(const int* __restrict__ src, const int* __restrict__ dst,
                               const float* __restrict__ e_buf,
                               const unsigned* __restrict__ m_enc,
                               const float* __restrict__ denom,
                               const float* __restrict__ feat,
                               float* __restrict__ z, int mp) {
  int lane = threadIdx.x & 31;
  int eid = (int)((blockIdx.x * (size_t)blockDim.x + threadIdx.x) >> 5);
  if (eid >= E_EDGES) return;                       // wave-uniform guard
  int s = src[eid], d = dst[eid];
  int c  = lane * 4;                                // 4 floats per lane
  int hh = c >> 5;                                  // head for this chunk
  float alpha = __expf(e_buf[eid * 4 + hh] - dec_f(m_enc[d * 4 + hh]))
                / denom[d * 4 + hh];
  const float4 f = *(const float4*)(feat + (size_t)s * HD + c);
  float* zp = z + (size_t)d * (MPATHS * HD) + mp * HD + c;
  atomicAdd(zp + 0, alpha * f.x);
  atomicAdd(zp + 1, alpha * f.y);
  atomicAdd(zp + 2, alpha * f.z);
  atomicAdd(zp + 3, alpha * f.w);
}

// ================= Kernel 6: ELU in place on z slice ======================
__global__ void elu_mp(float* __restrict__ z, int mp) {
  int idx = blockIdx.x * blockDim.x + threadIdx.x;
  if (idx >= N_NODES * HD) return;
  int n = idx >> 7, c = idx & 127;
  float* p = z + (size_t)n * (MPATHS * HD) + mp * HD + c;
  float v = *p;
  *p = v > 0.f ? v : (__expf(v) - 1.f);
}

// ===== Kernel 7: semantic scores via WMMA: sum_n tanh(z@w1+b1)@w2 per mp ==
// zflat is [N*M, 128]; row -> metapath = row % M. Only the per-mp mean of the
// projection is needed, so reduce straight into score[3].
__global__ void semantic_score(const float* __restrict__ z,
                               const float* __restrict__ w1,
                               const float* __restrict__ b1,
                               const float* __restrict__ w2,
                               float* __restrict__ score) {
  const int wave = threadIdx.x >> 5;
  const int lane = threadIdx.x & 31;
  const int rb = blockIdx.x * 8 + wave;             // 16-row block
  if (rb >= (N_NODES * MPATHS) / 16) return;        // wave-uniform guard
  const int r0 = rb * 16;
  const int hf = lane >> 4;
  const int l  = lane & 15;

  v8f acc[8];
  #pragma unroll
  for (int j = 0; j < 8; ++j) acc[j] = (v8f){0.f,0.f,0.f,0.f,0.f,0.f,0.f,0.f};

  const float* arow = z + (size_t)(r0 + l) * HD;
  for (int k0 = 0; k0 < HD; k0 += 4) {
    v2f a;
    a.x = arow[k0 + hf * 2 + 0];
    a.y = arow[k0 + hf * 2 + 1];
    #pragma unroll
    for (int j = 0; j < 8; ++j) {
      v2f b;
      b.x = w1[(size_t)(k0 + hf * 2 + 0) * HID_DIM + j * 16 + l];
      b.y = w1[(size_t)(k0 + hf * 2 + 1) * HID_DIM + j * 16 + l];
      acc[j] = __builtin_amdgcn_wmma_f32_16x16x4_f32(false, a, false, b,
                                                     (short)0, acc[j], false, false);
    }
  }
  // Epilogue: row = r0 + hf*8 + r ; col = j*16 + l. Reduce tanh(v+b1)*w2 over
  // cols (8 tiles in-register + 16-lane shuffle) -> one scalar per row.
  #pragma unroll
  for (int r = 0; r < 8; ++r) {
    float s = 0.f;
    #pragma unroll
    for (int j = 0; j < 8; ++j) {
      int col = j * 16 + l;
      s += tanhf(acc[j][r] + b1[col]) * w2[col];
    }
    s += __shfl_xor(s, 1, 32);
    s += __shfl_xor(s, 2, 32);
    s += __shfl_xor(s, 4, 32);
    s += __shfl_xor(s, 8, 32);
    if (l == 0) {
      int row = r0 + hf * 8 + r;
      atomicAdd(score + (row % MPATHS), s);
    }
  }
}

// ================= Kernel 8: beta = softmax(score / N) ====================
__global__ void beta_kernel(const float* __restrict__ score,
                            float* __restrict__ beta, float* __restrict__ out_att) {
  float s0 = score[0] / (float)N_NODES;
  float s1 = score[1] / (float)N_NODES;
  float s2 = score[2] / (float)N_NODES;
  float mx = fmaxf(s0, fmaxf(s1, s2));
  float e0 = __expf(s0 - mx), e1 = __expf(s1 - mx), e2 = __expf(s2 - mx);
  float inv = 1.f / (e0 + e1 + e2);
  beta[0] = e0 * inv; beta[1] = e1 * inv; beta[2] = e2 * inv;
  out_att[0] = beta[0]; out_att[1] = beta[1]; out_att[2] = beta[2];
}

// ================= Kernel 9: out = sum_m beta[m] * z[:,m,:] ===============
__global__ void combine(const float* __restrict__ z, const float* __restrict__ beta,
                        float* __restrict__ out) {
  int idx = blockIdx.x * blockDim.x + threadIdx.x;
  if (idx >= N_NODES * HD) return;
  int n = idx >> 7, c = idx & 127;
  const float* zp = z + (size_t)n * (MPATHS * HD) + c;
  out[idx] = beta[0] * zp[0] + beta[1] * zp[HD] + beta[2] * zp[2 * HD];
}

// ==========================================================================
extern "C" void kernel_launch(void* const* d_in, const int* in_sizes, int n_in,
                              void* d_out, int out_size, void* d_ws, size_t ws_size,
                              hipStream_t stream) {
  const float* h      = (const float*)d_in[0];
  const float* W      = (const float*)d_in[1];
  const float* attn_l = (const float*)d_in[2];
  const float* attn_r = (const float*)d_in[3];
  const float* w1     = (const float*)d_in[4];
  const float* b1     = (const float*)d_in[5];
  const float* w2     = (const float*)d_in[6];
  const int* srcs[MPATHS] = {(const int*)d_in[7], (const int*)d_in[9],  (const int*)d_in[11]};
  const int* dsts[MPATHS] = {(const int*)d_in[8], (const int*)d_in[10], (const int*)d_in[12]};
  float* out = (float*)d_out;

  float* ws = (float*)d_ws;
  float*    feat  = ws;             ws += (size_t)N_NODES * HD;
  float*    el    = ws;             ws += (size_t)N_NODES * HEADS;
  float*    er    = ws;             ws += (size_t)N_NODES * HEADS;
  float*    e_buf = ws;             ws += (size_t)E_EDGES * HEADS;
  unsigned* m_enc = (unsigned*)ws;  ws += (size_t)N_NODES * HEADS;
  float*    denom = ws;             ws += (size_t)N_NODES * HEADS;
  float*    z     = ws;             ws += (size_t)N_NODES * MPATHS * HD;
  float*    score = ws;             ws += 4;
  float*    beta  = ws;             ws += 4;

  // Shared projection (reference recomputes per metapath; h@W is identical).
  feat_gemm<<<N_NODES / 16, 256, 0, stream>>>(h, W, feat);
  el_er_kernel<<<(N_NODES * HEADS + 255) / 256, 256, 0, stream>>>(feat, attn_l, attn_r, el, er);
  zero_score<<<1, 32, 0, stream>>>(score);

  for (int mp = 0; mp < MPATHS; ++mp) {
    init_mp<<<(N_NODES * HD + 255) / 256, 256, 0, stream>>>(m_enc, denom, z, mp);
    edge_logits_max<<<(E_EDGES * HEADS + 255) / 256, 256, 0, stream>>>(
        srcs[mp], dsts[mp], el, er, e_buf, m_enc);
    edge_denom<<<(E_EDGES * HEADS + 255) / 256, 256, 0, stream>>>(
        dsts[mp], e_buf, m_enc, denom);
    edge_aggregate<<<(E_EDGES * 32 + 255) / 256, 256, 0, stream>>>(
        srcs[mp], dsts[mp], e_buf, m_enc, denom, feat, z, mp);
    elu_mp<<<(N_NODES * HD + 255) / 256, 256, 0, stream>>>(z, mp);
  }

  semantic_score<<<((N_NODES * MPATHS / 16) + 7) / 8, 256, 0, stream>>>(z, w1, b1, w2, score);
  beta_kernel<<<1, 1, 0, stream>>>(score, beta, out + (size_t)N_NODES * HD);
  combine<<<(N_NODES * HD + 255) / 256, 256, 0, stream>>>(z, beta, out);
}